// Model_Net_55155970015714
// MI455X (gfx1250) — compile-verified
//
#include <hip/hip_runtime.h>
#include <hip/hip_bf16.h>
#include <math.h>

#define LAYERS 3
#define NNODES 50000
#define NEDGES 500000
#define BATCH  131072
#define DIM    128
#define HID    64
#define BN_EPS 1e-5f

typedef __attribute__((ext_vector_type(2))) float v2f;
typedef __attribute__((ext_vector_type(8))) float v8f;

// ---------------------------------------------------------------------------
// Small precompute: Fd = Wg2@Wd [16,3], fd0 = bg2@Wd + bd [3],
//                   Fp = Wg2@Wp [16,2], fp0 = bg2@Wp [2]
// fused layout: [0..47]=Fd(j*3+c), [48..50]=fd0, [64..95]=Fp(j*2+c), [96..97]=fp0
// ---------------------------------------------------------------------------
__global__ void precompute_fused(const float* __restrict__ Wg2,
                                 const float* __restrict__ Wd,
                                 const float* __restrict__ bd,
                                 const float* __restrict__ Wp,
                                 const float* __restrict__ bg2,
                                 float* __restrict__ fused) {
    int t = threadIdx.x;
    if (t < 48) {
        int j = t / 3, c = t % 3;
        float s = 0.f;
        for (int k = 0; k < 256; ++k) s = fmaf(Wg2[j * 256 + k], Wd[k * 3 + c], s);
        fused[t] = s;
    } else if (t < 51) {
        int c = t - 48;
        float s = bd[c];
        for (int k = 0; k < 256; ++k) s = fmaf(bg2[k], Wd[k * 3 + c], s);
        fused[48 + c] = s;
    } else if (t < 83) {
        int i = t - 51;
        int j = i / 2, c = i % 2;
        float s = 0.f;
        for (int k = 0; k < 256; ++k) s = fmaf(Wg2[j * 256 + k], Wp[k * 2 + c], s);
        fused[64 + j * 2 + c] = s;
    } else if (t < 85) {
        int c = t - 83;
        float s = 0.f;
        for (int k = 0; k < 256; ++k) s = fmaf(bg2[k], Wp[k * 2 + c], s);
        fused[96 + c] = s;
    }
}

// ---------------------------------------------------------------------------
// Graph-side kernels
// ---------------------------------------------------------------------------
__global__ void deg_init(float* __restrict__ degf, int n) {
    int i = blockIdx.x * blockDim.x + threadIdx.x;
    if (i < n) degf[i] = 1.0f;  // self-loop
}

__global__ void deg_edges(const int* __restrict__ dst, float* __restrict__ degf, int e) {
    int i = blockIdx.x * blockDim.x + threadIdx.x;
    if (i < e) atomicAdd(&degf[dst[i]], 1.0f);
}

// dinv[n] = rsqrt(deg); ts[n] = x[n]*dinv^2 (self-loop term); zero stats
__global__ void dinv_ts_init(const float* __restrict__ degf, const float* __restrict__ x,
                             float* __restrict__ dinv, float* __restrict__ ts,
                             float* __restrict__ stats, int n) {
    int i = blockIdx.x * blockDim.x + threadIdx.x;
    if (i < n) {
        float d = rsqrtf(degf[i]);
        dinv[i] = d;
        ts[i] = x[i] * d * d;
    }
    if (blockIdx.x == 0 && threadIdx.x < 2) stats[threadIdx.x] = 0.f;
}

// ts[dst] += x[src] * dinv[src] * dinv[dst]  (scalar aggregation: conv1 is rank-1)
__global__ void t_edges(const int* __restrict__ src, const int* __restrict__ dst,
                        const float* __restrict__ x, const float* __restrict__ dinv,
                        float* __restrict__ ts, int e) {
    int i = blockIdx.x * blockDim.x + threadIdx.x;
    if (i < e) {
        int s = src[i], d = dst[i];
        atomicAdd(&ts[d], x[s] * dinv[s] * dinv[d]);
    }
}

// stats[0] += sum(t), stats[1] += sum(t^2)
__global__ void t_stats(const float* __restrict__ ts, float* __restrict__ stats, int n) {
    int i = blockIdx.x * blockDim.x + threadIdx.x;
    float v = (i < n) ? ts[i] : 0.f;
    float s = v, q = v * v;
    for (int off = 16; off > 0; off >>= 1) {
        s += __shfl_xor(s, off, 32);
        q += __shfl_xor(q, off, 32);
    }
    if ((threadIdx.x & 31) == 0) {
        atomicAdd(&stats[0], s);
        atomicAdd(&stats[1], q);
    }
}

// a_j = g_j*W1_j*rsqrt(var*W1_j^2+eps);  c_j = be_j - a_j*mean   (b1 cancels in BN)
__global__ void coef_kernel(const float* __restrict__ stats, const float* __restrict__ W1,
                            const float* __restrict__ g, const float* __restrict__ be,
                            float* __restrict__ ajbj, float inv_n) {
    int j = threadIdx.x;  // 64 threads
    float mean = stats[0] * inv_n;
    float var = stats[1] * inv_n - mean * mean;
    float w = W1[j];
    float a = g[j] * w * rsqrtf(var * w * w + BN_EPS);
    ajbj[j] = a;
    ajbj[64 + j] = be[j] - a * mean;
}

// ---------------------------------------------------------------------------
// conv2 GEMM with fp32 WMMA: z[N,128] = relu(a_j*t + c_j)[N,64] @ W2[64,128]
// A-fragments generated on the fly from t (conv1+BN+ReLU fused into A-load).
// One wave per 16-row tile; block = 8 waves = 128 rows. 3125 tiles exactly.
// ---------------------------------------------------------------------------
__global__ __launch_bounds__(256) void gemm_z(const float* __restrict__ ts,
                                              const float* __restrict__ ajbj,
                                              const float* __restrict__ W2,
                                              float* __restrict__ z, int nNodes) {
    __shared__ float sW2[HID * DIM];  // 32 KB
    __shared__ float saj[HID];
    __shared__ float scj[HID];
    __shared__ float st[128];

    int tid = threadIdx.x;
    for (int i = tid; i < HID * DIM; i += 256) sW2[i] = W2[i];
    if (tid < HID) {
        saj[tid] = ajbj[tid];
        scj[tid] = ajbj[HID + tid];
    }
    int row0 = blockIdx.x * 128;
    for (int i = tid; i < 128; i += 256) {
        int r = row0 + i;
        st[i] = (r < nNodes) ? ts[r] : 0.f;
    }
    __syncthreads();

    int wave = tid >> 5;
    int lane = tid & 31;
    int grow0 = row0 + wave * 16;
    if (grow0 >= nNodes) return;  // wave-uniform; EXEC stays all-1s for WMMA

    int half = lane >> 4;  // 0: K pair {0,1}, 1: K pair {2,3}
    int lm = lane & 15;
    float tm = st[wave * 16 + lm];

    v8f zero = {0.f, 0.f, 0.f, 0.f, 0.f, 0.f, 0.f, 0.f};
    v8f acc[8];
#pragma unroll
    for (int i = 0; i < 8; ++i) acc[i] = zero;

    for (int kk = 0; kk < HID; kk += 4) {
        int k0 = kk + half * 2;
        v2f a;
        a.x = fmaxf(fmaf(saj[k0], tm, scj[k0]), 0.f);
        a.y = fmaxf(fmaf(saj[k0 + 1], tm, scj[k0 + 1]), 0.f);
#pragma unroll
        for (int nt = 0; nt < 8; ++nt) {
            v2f b;
            b.x = sW2[k0 * DIM + nt * 16 + lm];
            b.y = sW2[(k0 + 1) * DIM + nt * 16 + lm];
            acc[nt] = __builtin_amdgcn_wmma_f32_16x16x4_f32(
                false, a, false, b, (short)0, acc[nt], false, false);
        }
    }

    int rbase = grow0 + half * 8;  // C layout: VGPR v -> row v (+8 for upper lanes)
#pragma unroll
    for (int nt = 0; nt < 8; ++nt) {
#pragma unroll
        for (int v = 0; v < 8; ++v) {
            z[(size_t)(rbase + v) * DIM + nt * 16 + lm] = acc[nt][v];
        }
    }
}

// emb_l[n][j] = z[n][j]*dinv[n]^2 + b2[j]   (self-loop + bias init before scatter)
__global__ void emb_init(const float* __restrict__ z, const float* __restrict__ dinv,
                         const float* __restrict__ b2, float* __restrict__ embl, int n) {
    int i = blockIdx.x * blockDim.x + threadIdx.x;
    if (i < n * DIM) {
        int node = i >> 7;
        int j = i & (DIM - 1);
        float d = dinv[node];
        embl[i] = fmaf(z[i], d * d, b2[j]);
    }
}

// emb_l[dst][:] += z[src][:] * dinv[src]*dinv[dst]  — one wave per edge:
// scalar src/dst/dinv loads, contiguous 512B vector traffic, float4 per lane.
__global__ __launch_bounds__(256) void agg_edges(const int* __restrict__ src,
                                                 const int* __restrict__ dst,
                                                 const float* __restrict__ z,
                                                 const float* __restrict__ dinv,
                                                 float* __restrict__ embl, int e) {
    int gid = blockIdx.x * blockDim.x + threadIdx.x;
    int eI = gid >> 5;  // wave-uniform
    if (eI >= e) return;
    int q = (gid & 31) << 2;
    int s = src[eI], d = dst[eI];
    float sc = dinv[s] * dinv[d];
    const float4 zv = *(const float4*)(z + (size_t)s * DIM + q);
    float* p = embl + (size_t)d * DIM + q;
    atomicAdd(p + 0, zv.x * sc);
    atomicAdd(p + 1, zv.y * sc);
    atomicAdd(p + 2, zv.z * sc);
    atomicAdd(p + 3, zv.w * sc);
}

// ---------------------------------------------------------------------------
// Batch MLP head: one thread per sample, streams the 256-wide gathered spec
// once; common(256) is algebraically eliminated via fused matrices.
// ---------------------------------------------------------------------------
__global__ __launch_bounds__(256) void batch_head(
    const float* __restrict__ emb, const int* __restrict__ leftnode,
    const int* __restrict__ rightnode, const int* __restrict__ layer,
    const float* __restrict__ Wg1, const float* __restrict__ bg1,
    const float* __restrict__ Ww, const float* __restrict__ bw,
    const float* __restrict__ Wp, const float* __restrict__ bp,
    const float* __restrict__ fused, float* __restrict__ out, int Bn) {
    int b = blockIdx.x * blockDim.x + threadIdx.x;
    if (b >= Bn) return;
    int l = layer[b], ln = leftnode[b], rn = rightnode[b];
    const float* pl = emb + ((size_t)l * NNODES + ln) * DIM;
    const float* pr = emb + ((size_t)l * NNODES + rn) * DIM;

    float s1[16];
#pragma unroll
    for (int j = 0; j < 16; ++j) s1[j] = 0.f;
    float lg0 = bw[0] + (float)l * Ww[0] + (float)ln * Ww[2] + (float)rn * Ww[4];
    float lg1 = bw[1] + (float)l * Ww[1] + (float)ln * Ww[3] + (float)rn * Ww[5];
    float sp0 = 0.f, sp1 = 0.f;

    for (int hlf = 0; hlf < 2; ++hlf) {
        const float* p = hlf ? pr : pl;
        int kb = hlf * 128;
        for (int k2 = 0; k2 < 128; ++k2) {
            float v = p[k2];
            int k = kb + k2;
            const float* wg = Wg1 + k * 16;  // uniform index -> s_load broadcast
#pragma unroll
            for (int j = 0; j < 16; ++j) s1[j] = fmaf(v, wg[j], s1[j]);
            lg0 = fmaf(v, Ww[(3 + k) * 2 + 0], lg0);
            lg1 = fmaf(v, Ww[(3 + k) * 2 + 1], lg1);
            sp0 = fmaf(v, Wp[k * 2 + 0], sp0);
            sp1 = fmaf(v, Wp[k * 2 + 1], sp1);
        }
    }

    const float* Fd = fused;
    const float* fd0 = fused + 48;
    const float* Fp = fused + 64;
    const float* fp0 = fused + 96;
    float d0 = fd0[0], d1 = fd0[1], d2 = fd0[2];
    float cp0 = fp0[0], cp1 = fp0[1];
#pragma unroll
    for (int j = 0; j < 16; ++j) {
        float r = fmaxf(s1[j] + bg1[j], 0.f);
        d0 = fmaf(r, Fd[j * 3 + 0], d0);
        d1 = fmaf(r, Fd[j * 3 + 1], d1);
        d2 = fmaf(r, Fd[j * 3 + 2], d2);
        cp0 = fmaf(r, Fp[j * 2 + 0], cp0);
        cp1 = fmaf(r, Fp[j * 2 + 1], cp1);
    }
    float m = fmaxf(lg0, lg1);
    float e0 = expf(lg0 - m), e1 = expf(lg1 - m);
    float inv = 1.f / (e0 + e1);
    float w0 = e0 * inv, w1 = e1 * inv;

    out[(size_t)b * 2 + 0] = w0 * sp0 + w1 * cp0 + bp[0];
    out[(size_t)b * 2 + 1] = w0 * sp1 + w1 * cp1 + bp[1];
    float* dout = out + (size_t)Bn * 2;
    dout[(size_t)b * 3 + 0] = d0;
    dout[(size_t)b * 3 + 1] = d1;
    dout[(size_t)b * 3 + 2] = d2;
}

// ---------------------------------------------------------------------------
extern "C" void kernel_launch(void* const* d_in, const int* in_sizes, int n_in,
                              void* d_out, int out_size, void* d_ws, size_t ws_size,
                              hipStream_t stream) {
    const float* x       = (const float*)d_in[0];
    const int* edge_idx  = (const int*)d_in[1];
    const int* leftnode  = (const int*)d_in[2];
    const int* rightnode = (const int*)d_in[3];
    const int* layer     = (const int*)d_in[4];
    const float* W1   = (const float*)d_in[5];
    // d_in[6] = b1 : cancels inside BatchNorm, unused
    const float* bn_g = (const float*)d_in[7];
    const float* bn_b = (const float*)d_in[8];
    const float* W2   = (const float*)d_in[9];
    const float* b2   = (const float*)d_in[10];
    const float* Wg1  = (const float*)d_in[11];
    const float* bg1  = (const float*)d_in[12];
    const float* Wg2  = (const float*)d_in[13];
    const float* bg2  = (const float*)d_in[14];
    const float* Ww   = (const float*)d_in[15];
    const float* bw   = (const float*)d_in[16];
    const float* Wd   = (const float*)d_in[17];
    const float* bd   = (const float*)d_in[18];
    const float* Wp   = (const float*)d_in[19];
    const float* bp   = (const float*)d_in[20];

    float* ws = (float*)d_ws;
    float* emb   = ws;                                     // L*N*DIM
    float* degf  = emb + (size_t)LAYERS * NNODES * DIM;    // N
    float* dinv  = degf + NNODES;                          // N
    float* ts    = dinv + NNODES;                          // N
    float* stats = ts + NNODES;                            // 8
    float* ajbj  = stats + 8;                              // 128
    float* fused = ajbj + 128;                             // 128
    float* zbuf  = fused + 128;                            // N*DIM

    const int NB = (NNODES + 255) / 256;
    const int EB = (NEDGES + 255) / 256;

    precompute_fused<<<1, 128, 0, stream>>>(Wg2, Wd, bd, Wp, bg2, fused);

    for (int l = 0; l < LAYERS; ++l) {
        const int* srcl = edge_idx + (size_t)l * 2 * NEDGES;
        const int* dstl = srcl + NEDGES;
        const float* xl = x + (size_t)l * NNODES;
        float* embl = emb + (size_t)l * NNODES * DIM;

        deg_init<<<NB, 256, 0, stream>>>(degf, NNODES);
        deg_edges<<<EB, 256, 0, stream>>>(dstl, degf, NEDGES);
        dinv_ts_init<<<NB, 256, 0, stream>>>(degf, xl, dinv, ts, stats, NNODES);
        t_edges<<<EB, 256, 0, stream>>>(srcl, dstl, xl, dinv, ts, NEDGES);
        t_stats<<<NB, 256, 0, stream>>>(ts, stats, NNODES);
        coef_kernel<<<1, 64, 0, stream>>>(stats, W1 + (size_t)l * HID,
                                          bn_g + (size_t)l * HID, bn_b + (size_t)l * HID,
                                          ajbj, 1.0f / (float)NNODES);
        gemm_z<<<(3125 + 7) / 8, 256, 0, stream>>>(ts, ajbj, W2 + (size_t)l * HID * DIM,
                                                   zbuf, NNODES);
        emb_init<<<(NNODES * DIM + 255) / 256, 256, 0, stream>>>(zbuf, dinv,
                                                                 b2 + (size_t)l * DIM,
                                                                 embl, NNODES);
        agg_edges<<<(NEDGES * 32) / 256, 256, 0, stream>>>(srcl, dstl, zbuf, dinv,
                                                           embl, NEDGES);
    }

    batch_head<<<BATCH / 256, 256, 0, stream>>>(emb, leftnode, rightnode, layer,
                                                Wg1, bg1, Ww, bw, Wp, bp, fused,
                                                (float*)d_out, BATCH);
}